// GNN_LEP_541165879466
// MI455X (gfx1250) — compile-verified
//
#include <hip/hip_runtime.h>

typedef __attribute__((ext_vector_type(2))) float v2f;
typedef __attribute__((ext_vector_type(8))) float v8f;

// ---------------------------------------------------------------------------
// WMMA GEMM: Y[M,K] = X[M,F] @ W[F,K]   (fp32, V_WMMA_F32_16X16X4_F32)
// One wave computes a full 16-row x K strip: NT = K/16 accumulators, one A
// fragment shared across NT WMMAs per k-step. W is staged in LDS as
// (row-pair, col) float2 so each B fragment is one ds_load_b64.
//
// A frag (16x4): lane<16 -> row m0+l, k..k+1 ; lane>=16 -> row m0+l, k+2..k+3
// B frag (4x16): lane<16 -> col n0+l, rows k,k+1 ; lane>=16 -> rows k+2,k+3
// C/D: VGPR i -> row m0+i+8*(lane>=16), col n0+(lane&15)
// ---------------------------------------------------------------------------
template <int F, int K>
__global__ void hg_wmma_gemm(const float* __restrict__ X,
                             const float* __restrict__ W,
                             float* __restrict__ Y, int M) {
  constexpr int NT = K / 16;                 // 16x16 tiles along the out dim
  __shared__ v2f Wl[(F / 2) * K];            // Wl[p*K+n] = {W[2p][n], W[2p+1][n]}

  // Cooperative staging of W into LDS (32 KB), interleaved by row pairs.
  for (int idx = threadIdx.x; idx < (F / 2) * K; idx += blockDim.x) {
    const int p = idx / K;
    const int n = idx - p * K;
    v2f t;
    t.x = W[(size_t)(2 * p) * K + n];
    t.y = W[(size_t)(2 * p + 1) * K + n];
    Wl[idx] = t;
  }
  __syncthreads();

  const int wave = (blockIdx.x * blockDim.x + threadIdx.x) >> 5;
  const int lane = threadIdx.x & 31;
  const int mtiles = (M + 15) >> 4;

  if (wave < mtiles) {                       // whole-wave guard: EXEC all-1 inside
    const int m0 = wave << 4;
    const int half = lane >> 4;              // 0 or 1
    const int l = lane & 15;

    int arow = m0 + l;
    if (arow >= M) arow = M - 1;             // clamp (no exec divergence; padding
                                             // rows computed but never stored)
    const float* Xrow = X + (size_t)arow * F + 2 * half;

    v8f acc[NT] = {};
#pragma unroll 2
    for (int k = 0; k < F; k += 4) {
      const v2f a = *(const v2f*)(Xrow + k); // 8B-aligned: F%4==0, k%4==0
      const v2f* brow = &Wl[(size_t)(k / 2 + half) * K + l];
#pragma unroll
      for (int t = 0; t < NT; ++t) {
        const v2f b = brow[t * 16];          // ds_load_b64 (2addr-paired)
        acc[t] = __builtin_amdgcn_wmma_f32_16x16x4_f32(
            /*neg_a=*/false, a, /*neg_b=*/false, b,
            /*c_mod=*/(short)0, acc[t], /*reuse_a=*/false, /*reuse_b=*/false);
      }
    }

    // Epilogue: wave-uniform bound check -> branch-free coalesced stores on
    // the fast path (every strip when M % 16 == 0), guarded only on a ragged
    // final strip.
    float* Yb = Y + (size_t)(m0 + half * 8) * K + l;
    if (m0 + 16 <= M) {
#pragma unroll
      for (int t = 0; t < NT; ++t) {
#pragma unroll
        for (int i = 0; i < 8; ++i) {
          Yb[(size_t)i * K + t * 16] = acc[t][i];
        }
      }
    } else {
#pragma unroll
      for (int t = 0; t < NT; ++t) {
#pragma unroll
        for (int i = 0; i < 8; ++i) {
          if (m0 + half * 8 + i < M) Yb[(size_t)i * K + t * 16] = acc[t][i];
        }
      }
    }
  }
}

// ---------------------------------------------------------------------------
// Degree accumulation: D[node] += w[edge]; Bdeg[edge] += 1
// ---------------------------------------------------------------------------
__global__ void hg_degrees(const int* __restrict__ node_idx,
                           const int* __restrict__ edge_idx,
                           const float* __restrict__ w,
                           float* __restrict__ D, float* __restrict__ Bdeg,
                           int E) {
  const int i = blockIdx.x * blockDim.x + threadIdx.x;
  if (i >= E) return;
  const int e = edge_idx[i];
  atomicAdd(&D[node_idx[i]], w[e]);
  atomicAdd(&Bdeg[e], 1.0f);
}

// In-place invert: v = (v > 0) ? 1/v : 0  for both D (len N) and Bdeg (len NH)
__global__ void hg_invert(float* __restrict__ D, int N,
                          float* __restrict__ B, int NH) {
  const int i = blockIdx.x * blockDim.x + threadIdx.x;
  if (i < N) {
    const float d = D[i];
    D[i] = d > 0.0f ? 1.0f / d : 0.0f;
  }
  if (i < NH) {
    const float b = B[i];
    B[i] = b > 0.0f ? 1.0f / b : 0.0f;
  }
}

// ---------------------------------------------------------------------------
// Scatter node features into hyperedges: e[edge_idx[i]] += xw[node_idx[i]]
// K/4 threads per incidence, float4 gathers + 4 fp32 global atomics.
// ---------------------------------------------------------------------------
template <int K>
__global__ void hg_scatter_n2e(const float* __restrict__ xw,
                               const int* __restrict__ node_idx,
                               const int* __restrict__ edge_idx,
                               float* __restrict__ e, int E) {
  constexpr int CPI = K / 4;
  const int t = blockIdx.x * blockDim.x + threadIdx.x;
  const int i = t / CPI;
  const int c = t - i * CPI;
  if (i >= E) return;
  const float4 v = ((const float4*)(xw + (size_t)node_idx[i] * K))[c];
  float* dst = e + (size_t)edge_idx[i] * K + c * 4;
  atomicAdd(dst + 0, v.x);
  atomicAdd(dst + 1, v.y);
  atomicAdd(dst + 2, v.z);
  atomicAdd(dst + 3, v.w);
}

// Scatter hyperedge features back to nodes: out[node_idx[i]] += e[edge_idx[i]]
template <int K>
__global__ void hg_scatter_e2n(const float* __restrict__ e,
                               const int* __restrict__ node_idx,
                               const int* __restrict__ edge_idx,
                               float* __restrict__ out, int E) {
  constexpr int CPI = K / 4;
  const int t = blockIdx.x * blockDim.x + threadIdx.x;
  const int i = t / CPI;
  const int c = t - i * CPI;
  if (i >= E) return;
  const float4 v = ((const float4*)(e + (size_t)edge_idx[i] * K))[c];
  float* dst = out + (size_t)node_idx[i] * K + c * 4;
  atomicAdd(dst + 0, v.x);
  atomicAdd(dst + 1, v.y);
  atomicAdd(dst + 2, v.z);
  atomicAdd(dst + 3, v.w);
}

// e[h, :] *= Binv[h] * w[h]   (fused B^-1 and edge-weight scaling)
template <int K>
__global__ void hg_scale_e(float* __restrict__ e,
                           const float* __restrict__ Binv,
                           const float* __restrict__ w, int NH) {
  constexpr int CPI = K / 4;
  const int t = blockIdx.x * blockDim.x + threadIdx.x;
  const int h = t / CPI;
  if (h >= NH) return;
  const float s = Binv[h] * w[h];
  float4 v = ((float4*)e)[t];
  v.x *= s; v.y *= s; v.z *= s; v.w *= s;
  ((float4*)e)[t] = v;
}

// out[n, :] = relu(out[n, :] * Dinv[n] + bias)
template <int K>
__global__ void hg_finalize(float* __restrict__ out,
                            const float* __restrict__ Dinv,
                            const float* __restrict__ bias, int N) {
  constexpr int CPI = K / 4;
  const int t = blockIdx.x * blockDim.x + threadIdx.x;
  const int n = t / CPI;
  const int c = t - n * CPI;
  if (n >= N) return;
  const float s = Dinv[n];
  const float4 bb = ((const float4*)bias)[c];
  float4 v = ((float4*)out)[t];
  v.x = fmaxf(fmaf(v.x, s, bb.x), 0.0f);
  v.y = fmaxf(fmaf(v.y, s, bb.y), 0.0f);
  v.z = fmaxf(fmaf(v.z, s, bb.z), 0.0f);
  v.w = fmaxf(fmaf(v.w, s, bb.w), 0.0f);
  ((float4*)out)[t] = v;
}

static inline int cdiv(long long a, int b) { return (int)((a + b - 1) / b); }

extern "C" void kernel_launch(void* const* d_in, const int* in_sizes, int n_in,
                              void* d_out, int out_size, void* d_ws, size_t ws_size,
                              hipStream_t stream) {
  constexpr int F_IN = 128, H1 = 64, H2 = 128;

  const float* x  = (const float*)d_in[0];
  const int*   ei = (const int*)d_in[1];    // [2, E]: row0 = node_idx, row1 = edge_idx
  const float* w  = (const float*)d_in[2];  // [NH]
  // d_in[3] = batch (int64), unused in forward
  const float* W1 = (const float*)d_in[4];
  const float* b1 = (const float*)d_in[5];
  const float* W2 = (const float*)d_in[6];
  const float* b2 = (const float*)d_in[7];

  const int N  = in_sizes[0] / F_IN;
  const int E  = in_sizes[1] / 2;
  const int NH = in_sizes[2];
  const int* node_idx = ei;
  const int* edge_idx = ei + E;

  // ---- workspace carve-up (256B aligned), ~181 MB total ----
  char* ws = (char*)d_ws;
  size_t off = 0;
  auto carve = [&](size_t bytes) -> void* {
    void* p = ws + off;
    off = (off + bytes + 255) & ~(size_t)255;
    return p;
  };
  float* Dinv = (float*)carve((size_t)N * 4);            // node degree -> inverse
  float* Binv = (float*)carve((size_t)NH * 4);           // hyperedge degree -> inverse
  float* xw   = (float*)carve((size_t)N * H2 * 4);       // GEMM output (max K=128)
  float* ebuf = (float*)carve((size_t)NH * H2 * 4);      // hyperedge features (max K=128)
  float* h    = (float*)carve((size_t)N * H1 * 4);       // layer-1 output
  (void)ws_size;

  const int B = 256;

  // ---- degrees (shared by both layers) ----
  hipMemsetAsync(Dinv, 0, (size_t)N * 4, stream);
  hipMemsetAsync(Binv, 0, (size_t)NH * 4, stream);
  hg_degrees<<<cdiv(E, B), B, 0, stream>>>(node_idx, edge_idx, w, Dinv, Binv, E);
  hg_invert<<<cdiv(NH > N ? NH : N, B), B, 0, stream>>>(Dinv, N, Binv, NH);

  // ================= Layer 1: F=128 -> K=64 =================
  {
    constexpr int K = H1;
    const long long waves = (N + 15) / 16;   // one wave per 16-row strip
    hg_wmma_gemm<F_IN, K><<<cdiv(waves * 32, B), B, 0, stream>>>(x, W1, xw, N);

    hipMemsetAsync(ebuf, 0, (size_t)NH * K * 4, stream);
    hg_scatter_n2e<K><<<cdiv((long long)E * (K / 4), B), B, 0, stream>>>(
        xw, node_idx, edge_idx, ebuf, E);
    hg_scale_e<K><<<cdiv((long long)NH * (K / 4), B), B, 0, stream>>>(ebuf, Binv, w, NH);

    hipMemsetAsync(h, 0, (size_t)N * K * 4, stream);
    hg_scatter_e2n<K><<<cdiv((long long)E * (K / 4), B), B, 0, stream>>>(
        ebuf, node_idx, edge_idx, h, E);
    hg_finalize<K><<<cdiv((long long)N * (K / 4), B), B, 0, stream>>>(h, Dinv, b1, N);
  }

  // ================= Layer 2: F=64 -> K=128 =================
  {
    constexpr int K = H2;
    float* out = (float*)d_out;
    const long long waves = (N + 15) / 16;   // one wave per 16-row strip
    hg_wmma_gemm<H1, K><<<cdiv(waves * 32, B), B, 0, stream>>>(h, W2, xw, N);

    hipMemsetAsync(ebuf, 0, (size_t)NH * K * 4, stream);
    hg_scatter_n2e<K><<<cdiv((long long)E * (K / 4), B), B, 0, stream>>>(
        xw, node_idx, edge_idx, ebuf, E);
    hg_scale_e<K><<<cdiv((long long)NH * (K / 4), B), B, 0, stream>>>(ebuf, Binv, w, NH);

    hipMemsetAsync(out, 0, (size_t)N * K * 4, stream);
    hg_scatter_e2n<K><<<cdiv((long long)E * (K / 4), B), B, 0, stream>>>(
        ebuf, node_idx, edge_idx, out, E);
    hg_finalize<K><<<cdiv((long long)N * (K / 4), B), B, 0, stream>>>(out, Dinv, b2, N);
  }
}